// FusionLoss_3667902071379
// MI455X (gfx1250) — compile-verified
//
#include <hip/hip_runtime.h>

// -------------------------------------------------------------------------
// FusionLoss for MI455X (gfx1250, wave32).
//
// Work: B=256 samples, K=16 keypoints, 12 limb edges, 4 groups.
//   reg  = smooth_l1((gather(output, ind) - target) * mask) / (sum(mask)+1e-4)
//   var  = mean over samples with mask.sum()==0 of the per-group limb-length
//          variance term
//   loss = reg + 0.01 * var
//
// Traffic is ~0.6 MB total -> pure latency-bound; one 256-thread block
// (8 wave32 waves), one thread per sample, everything in registers.
// Final 32-lane reduction goes through V_WMMA_F32_16X16X4_F32 with
// A = ones(16x4): D[m][n] = v[n] + v[n+16] exactly (f32 FMA path), then
// 4 xor-shuffles finish the wave sum. Cross-wave via LDS.
// -------------------------------------------------------------------------

typedef float v2f __attribute__((ext_vector_type(2)));
typedef float v8f __attribute__((ext_vector_type(8)));

#define K_PTS 16

// Exact f32 sum of x across the 32 lanes of a wave, using the matrix pipe
// for the first fold. Requires EXEC == all ones (call from convergent code).
__device__ __forceinline__ float wave_sum_f32(float x) {
  v2f a; a[0] = 1.0f; a[1] = 1.0f;   // A(16x4) = all ones (both K-VGPRs)
  v2f b; b[0] = x;    b[1] = 0.0f;   // B(4x16): rows in VGPR0 = lane data
  v8f c = {};
  // 8 args: (neg_a, A, neg_b, B, c_mod, C, reuse_a, reuse_b)
  c = __builtin_amdgcn_wmma_f32_16x16x4_f32(
      false, a, false, b, (short)0, c, false, false);
  // Every lane: c[0] = v[lane&15] + v[(lane&15)+16]  (all D rows identical)
  float s = c[0];
  s += __shfl_xor(s, 1, 32);
  s += __shfl_xor(s, 2, 32);
  s += __shfl_xor(s, 4, 32);
  s += __shfl_xor(s, 8, 32);
  return s;   // full 32-lane sum in every lane
}

__global__ __launch_bounds__(256)
void fusion_loss_kernel(const float* __restrict__ output,  // (B,1,H,W) flat
                        const int*   __restrict__ mask,    // (B,16)
                        const int*   __restrict__ ind,     // (B,16)
                        const float* __restrict__ target,  // (B,16,1)
                        const float* __restrict__ gt2d,    // (B,32)
                        float*       __restrict__ out,     // scalar
                        int B, int HW)
{
  const int b = threadIdx.x;

  // Edge tables (compile-time folded under full unroll).
  const int   id1[12] = {0, 1, 3, 4, 10, 11, 13, 14, 2, 3, 12, 13};
  const int   id2[12] = {1, 2, 4, 5, 11, 12, 14, 15, 6, 6,  8,  8};
  const float wgt[12] = {1.0085885098415446f, 1.0f, 1.0f, 1.0085885098415446f,
                         1.1375361376887123f, 1.0f, 1.0f, 1.1375361376887123f,
                         1.0f, 1.0f, 1.0f, 1.0f};
  const int   gid[12] = {0, 0, 0, 0, 1, 1, 1, 1, 2, 2, 3, 3};

  // ---- per-sample loads (coalesced across threads) ----
  int   mk[K_PTS], ik[K_PTS];
  float tg[K_PTS], pred[K_PTS], xy[2 * K_PTS];

  const int*   mrow = mask   + b * K_PTS;
  const int*   irow = ind    + b * K_PTS;
  const float* trow = target + b * K_PTS;
  const float* grow = gt2d   + b * 2 * K_PTS;
  const float* orow = output + (size_t)b * (size_t)HW;

#pragma unroll
  for (int k = 0; k < K_PTS; ++k) { mk[k] = mrow[k]; ik[k] = irow[k]; tg[k] = trow[k]; }
#pragma unroll
  for (int j = 0; j < 2 * K_PTS; ++j) { xy[j] = grow[j]; }
  // feat[b,p,0] == output_flat[b*HW + p] (channel dim is 1, transpose is a no-op)
#pragma unroll
  for (int k = 0; k < K_PTS; ++k) { pred[k] = orow[ik[k]]; }

  // ---- regression term: masked smooth-L1 ----
  int   msum = 0;
  float sl1  = 0.0f;
#pragma unroll
  for (int k = 0; k < K_PTS; ++k) {
    msum += mk[k];
    float d  = (pred[k] - tg[k]) * (float)mk[k];
    float ad = fabsf(d);
    sl1 += (ad < 1.0f) ? (0.5f * ad * ad) : (ad - 0.5f);
  }

  // ---- limb-variance term (branch-free: keep EXEC full for the WMMA) ----
  float l[12];
  int   vis[12];
  float num_g[4] = {0.f, 0.f, 0.f, 0.f};
  float sum_g[4] = {0.f, 0.f, 0.f, 0.f};
#pragma unroll
  for (int e = 0; e < 12; ++e) {
    const int i1 = id1[e], i2 = id2[e];
    float dx = xy[2 * i1]     - xy[2 * i2];
    float dy = xy[2 * i1 + 1] - xy[2 * i2 + 1];
    float dz = pred[i1]       - pred[i2];
    l[e]   = sqrtf(dx * dx + dy * dy + dz * dz) * wgt[e];
    vis[e] = (tg[i1] > 0.5f) && (tg[i2] > 0.5f);
    float vf = vis[e] ? 1.0f : 0.0f;
    num_g[gid[e]] += vf;
    sum_g[gid[e]] += l[e] * vf;
  }
  float E[4], NG[4];
#pragma unroll
  for (int g = 0; g < 4; ++g) {
    float ngm = fmaxf(num_g[g], 1.0f);
    E[g]  = (num_g[g] > 0.5f) ? (sum_g[g] / ngm) : 0.0f;
    NG[g] = ngm;
  }
  float per_sample = 0.0f;
#pragma unroll
  for (int e = 0; e < 12; ++e) {
    float d    = l[e] - E[gid[e]];
    float term = d * d / (2.0f * NG[gid[e]]);
    per_sample += (vis[e] && (l[e] > 0.0f)) ? term : 0.0f;
  }
  float pvar = (msum == 0) ? per_sample : 0.0f;

  // ---- reduction: wave32 via WMMA+shuffle, then LDS across the 8 waves ----
  float t_sl1 = wave_sum_f32(sl1);
  float t_num = wave_sum_f32((float)msum);
  float t_var = wave_sum_f32(pvar);

  __shared__ float red[3][8];
  const int wave = threadIdx.x >> 5;
  const int lane = threadIdx.x & 31;
  if (lane == 0) { red[0][wave] = t_sl1; red[1][wave] = t_num; red[2][wave] = t_var; }
  __syncthreads();

  if (threadIdx.x == 0) {
    float a = 0.f, n = 0.f, v = 0.f;
    const int nw = blockDim.x >> 5;
#pragma unroll
    for (int w = 0; w < 8; ++w) {
      if (w < nw) { a += red[0][w]; n += red[1][w]; v += red[2][w]; }
    }
    float reg = a / (n + 0.0001f);
    float var = v / (float)B;
    out[0] = 1.0f * reg + 0.01f * var;   // REG_WEIGHT=1.0, VAR_WEIGHT=0.01
  }
}

extern "C" void kernel_launch(void* const* d_in, const int* in_sizes, int n_in,
                              void* d_out, int out_size, void* d_ws, size_t ws_size,
                              hipStream_t stream) {
  const float* output = (const float*)d_in[0];   // (B,1,H,W) f32
  const int*   mask   = (const int*)  d_in[1];   // (B,16) i32
  const int*   ind    = (const int*)  d_in[2];   // (B,16) i32
  const float* target = (const float*)d_in[3];   // (B,16,1) f32
  const float* gt2d   = (const float*)d_in[4];   // (B,32) f32
  float*       out    = (float*)d_out;

  const int B  = in_sizes[1] / K_PTS;            // 256
  const int HW = in_sizes[0] / B;                // 65536

  // One block, one thread per sample (B=256 -> 8 wave32 waves on one WGP).
  fusion_loss_kernel<<<1, B, 0, stream>>>(output, mask, ind, target, gt2d,
                                          out, B, HW);
}